// rnn_test_60481729462992
// MI455X (gfx1250) — compile-verified
//
#include <hip/hip_runtime.h>
#include <math.h>

// ---------------- problem constants ----------------
#define B_    256
#define T_    250
#define IN_   700
#define NH_   128
#define BR_   8
#define NO_   20
#define INS_  828          // 700 + 128
#define KBT_  207          // 828/4 K-blocks total
#define KBI_  175          // input K-blocks (k < 700)
#define KBR_  32           // recurrent K-blocks (k >= 700)
#define TC_   25           // timesteps per chunk
#define NCH_  10           // chunks (10 * 25 = 250)
#define VTH_  1.0f

// ---------------- workspace layout (float offsets) ----------------
static constexpr size_t OFF_BP    = 0;                           // packed (W1*mask)^T : 207*1024*4
static constexpr size_t NBP       = (size_t)KBT_ * 1024 * 4;     // 847,872
static constexpr size_t OFF_XP    = OFF_BP + NBP;                // xproj chunk: 25*256*1024
static constexpr size_t NXP       = (size_t)TC_ * B_ * 1024;     // 6,553,600
static constexpr size_t OFF_DST   = OFF_XP + NXP;                // dstate 256*1024
static constexpr size_t OFF_M1    = OFF_DST + (size_t)B_ * 1024; // mem1 256*128
static constexpr size_t OFF_SP    = OFF_M1 + (size_t)B_ * NH_;   // sp1  256*128
static constexpr size_t OFF_APRE  = OFF_SP + (size_t)B_ * NH_;   // accPre
static constexpr size_t OFF_M2    = OFF_APRE + (size_t)B_ * NH_; // mem2 256*20
static constexpr size_t OFF_OUT   = OFF_M2 + (size_t)B_ * NO_;   // out accumulator
static constexpr size_t OFF_APOST = OFF_OUT + (size_t)B_ * NO_;  // accPost
static constexpr size_t OFF_KN1   = OFF_APOST + (size_t)B_ * NO_;// sigmoid(tau_n1) [1024]
static constexpr size_t OFF_KM1   = OFF_KN1 + 1024;              // sigmoid(tau_m1) [128]
static constexpr size_t OFF_KM2   = OFF_KM1 + NH_;               // sigmoid(tau_m2) [20]

// ---------------- WMMA types ----------------
typedef float v2f __attribute__((ext_vector_type(2)));
typedef float v8f __attribute__((ext_vector_type(8)));

__device__ __forceinline__ v8f wmma_f32(v2f a, v2f b, v8f c) {
  // D = A(16x4) * B(4x16) + C(16x16), fp32, wave32
  return __builtin_amdgcn_wmma_f32_16x16x4_f32(false, a, false, b, (short)0, c, false, false);
}

__device__ __forceinline__ float sigm(float x) { return 1.0f / (1.0f + expf(-x)); }

// =====================================================================
// Kernel 0: pack B matrix (W1*mask)^T into WMMA-friendly blocks + init state
//   Bp[kb][n][kk] = W1[n][kb*4+kk] * mask[n][kb*4+kk]
// =====================================================================
__global__ __launch_bounds__(256) void k_init(const float* __restrict__ W1,
                                              const float* __restrict__ mask,
                                              const float* __restrict__ tau_m1,
                                              const float* __restrict__ tau_n1,
                                              const float* __restrict__ tau_m2,
                                              const float* __restrict__ mem0,
                                              float* __restrict__ ws) {
  size_t idx = (size_t)blockIdx.x * 256 + threadIdx.x;
  if (idx < NBP) {
    int kb = (int)(idx >> 12);
    int rem = (int)(idx & 4095);
    int n = rem >> 2, kk = rem & 3;
    int k = kb * 4 + kk;                        // k < 828 always
    ws[OFF_BP + idx] = W1[(size_t)n * INS_ + k] * mask[(size_t)n * INS_ + k];
  }
  if (idx < (size_t)B_ * 1024) ws[OFF_DST + idx] = 0.0f;
  if (idx < (size_t)B_ * NH_) {
    ws[OFF_M1 + idx]   = mem0[idx];
    ws[OFF_SP + idx]   = 0.0f;
    ws[OFF_APRE + idx] = 0.0f;
  }
  if (idx < (size_t)B_ * NO_) {
    ws[OFF_M2 + idx]    = 0.0f;
    ws[OFF_OUT + idx]   = 0.0f;
    ws[OFF_APOST + idx] = 0.0f;
  }
  if (idx < 1024) ws[OFF_KN1 + idx] = sigm(tau_n1[idx]);
  if (idx < NH_)  ws[OFF_KM1 + idx] = sigm(tau_m1[idx]);
  if (idx < NO_)  ws[OFF_KM2 + idx] = sigm(tau_m2[idx]);
}

// =====================================================================
// Kernel 1: xproj[tl][b][j] = x[b, t0+tl, :] @ Wm[:, :700].T + b1   (per chunk)
//   grid = (400 M-tiles, 4 N-groups), block = 256 (8 waves, wave32)
//   Each wave computes TWO 16x16 output tiles (shares the A operand).
//   A tile (16 x 700) staged to LDS via async-to-LDS; software-pipelined
//   depth-5 register double-buffering so WMMA never drains the counters.
// =====================================================================
#define LDA_ 708   // LDS row stride (pad: 708 % 64 == 4 -> conflict-free ds_load_b64)

__global__ __launch_bounds__(256) void k_xproj(const float* __restrict__ x,
                                               const float* __restrict__ b1,
                                               float* __restrict__ ws, int t0) {
  __shared__ float As[16 * LDA_];               // 45,312 B
  const float* Bp = ws + OFF_BP;
  float* xproj = ws + OFF_XP;

  int tid = threadIdx.x;
  int mt = blockIdx.x, ng = blockIdx.y;
  int tl = mt >> 4;                             // local timestep in chunk
  int b0 = (mt & 15) << 4;                      // batch-row base of this 16-row tile
  int t  = t0 + tl;

  // ---- async stage of x[b0..b0+15, t, 0:700] into LDS (16B granules) ----
  unsigned long long xbase = (unsigned long long)(size_t)x;
  for (int i = tid; i < 16 * (IN_ / 4); i += 256) {   // 2800 float4 transfers
    int r = i / (IN_ / 4);
    int kq = i - r * (IN_ / 4);
    unsigned voff = (unsigned)(((((size_t)(b0 + r) * T_) + t) * IN_ + (size_t)kq * 4) * 4);
    unsigned laddr = (unsigned)(size_t)&As[r * LDA_ + kq * 4];
    asm volatile("global_load_async_to_lds_b128 %0, %1, %2"
                 :: "v"(laddr), "v"(voff), "s"(xbase) : "memory");
  }
  asm volatile("s_wait_asynccnt 0" ::: "memory");
  __syncthreads();

  int lane = tid & 31, wave = tid >> 5;
  int col = lane & 15, kh = lane >> 4;
  int n0 = ng * 256 + wave * 32;                // two adjacent 16-wide tiles

  v8f c0, c1;
  float bb0 = b1[n0 + col];
  float bb1 = b1[n0 + 16 + col];
  #pragma unroll
  for (int v = 0; v < 8; ++v) { c0[v] = bb0; c1[v] = bb1; }

  const float* bp0 = Bp + (size_t)(n0 + col) * 4 + kh * 2;
  const float* bp1 = bp0 + 64;                  // +16 columns
  const float* ap  = &As[col * LDA_ + kh * 2];

  // ---- depth-5 software pipeline over 175 K-blocks (175 = 35 * 5) ----
  v2f a[5], bA[5], bB[5];
  #pragma unroll
  for (int u = 0; u < 5; ++u) {
    a[u]  = *(const v2f*)(ap + u * 4);
    bA[u] = *(const v2f*)(bp0 + (size_t)u * 4096);
    bB[u] = *(const v2f*)(bp1 + (size_t)u * 4096);
  }
  for (int g = 0; g < 34; ++g) {                // groups 0..33 reload, last group below
    int base = g * 5 + 5;
    #pragma unroll
    for (int u = 0; u < 5; ++u) {
      c0 = wmma_f32(a[u], bA[u], c0);
      c1 = wmma_f32(a[u], bB[u], c1);
      a[u]  = *(const v2f*)(ap + (base + u) * 4);
      bA[u] = *(const v2f*)(bp0 + (size_t)(base + u) * 4096);
      bB[u] = *(const v2f*)(bp1 + (size_t)(base + u) * 4096);
    }
  }
  #pragma unroll
  for (int u = 0; u < 5; ++u) {                 // epilogue: kb = 170..174
    c0 = wmma_f32(a[u], bA[u], c0);
    c1 = wmma_f32(a[u], bB[u], c1);
  }

  float* xp = xproj + ((size_t)(tl * B_ + b0 + 8 * kh)) * 1024 + n0 + col;
  #pragma unroll
  for (int v = 0; v < 8; ++v) {
    xp[(size_t)v * 1024]      = c0[v];
    xp[(size_t)v * 1024 + 16] = c1[v];
  }
}

// =====================================================================
// Kernel 2: recurrent chunk. 16 independent workgroups, one per 16 batch rows.
//   dstate lives in 64 accumulator VGPRs per lane across all 25 steps.
//   Recurrent GEMM: one kb-loop drives all 8 n-tiles -> 1 ds_load + 8
//   global loads per 8 back-to-back WMMAs, double-buffered.
// =====================================================================
#define LSP_ 132   // sp1 LDS row stride (132 % 64 == 4 -> conflict-free)

__global__ __launch_bounds__(256) void k_recur(const float* __restrict__ W2,
                                               const float* __restrict__ b2,
                                               float* __restrict__ ws,
                                               float* __restrict__ dout,
                                               int t0, int isLast) {
  __shared__ float sp1b[2][16 * LSP_];
  __shared__ float mem1s[16 * NH_];
  __shared__ float apres[16 * NH_];
  __shared__ float W2s[NO_ * NH_];
  __shared__ float mem2s[16 * NO_], outs[16 * NO_], aposts[16 * NO_];

  int tid = threadIdx.x;
  int b0 = blockIdx.x * 16;

  // ---- load persistent state ----
  for (int i = tid; i < 16 * NH_; i += 256) {
    int r = i >> 7, n = i & 127;
    mem1s[i] = ws[OFF_M1 + (size_t)(b0 + r) * NH_ + n];
    apres[i] = ws[OFF_APRE + (size_t)(b0 + r) * NH_ + n];
    sp1b[0][r * LSP_ + n] = ws[OFF_SP + (size_t)(b0 + r) * NH_ + n];
  }
  for (int i = tid; i < NO_ * NH_; i += 256) W2s[i] = W2[i];
  for (int i = tid; i < 16 * NO_; i += 256) {
    int r = i / NO_, o = i - r * NO_;
    mem2s[i]  = ws[OFF_M2 + (size_t)(b0 + r) * NO_ + o];
    outs[i]   = ws[OFF_OUT + (size_t)(b0 + r) * NO_ + o];
    aposts[i] = ws[OFF_APOST + (size_t)(b0 + r) * NO_ + o];
  }

  int lane = tid & 31, wave = tid >> 5;
  int col = lane & 15, kh = lane >> 4;

  // ---- per-lane constants + dstate in VGPRs (8 n-tiles x v8f) ----
  v8f dst[8];
  float kn1v[8], km1v[8];
  #pragma unroll
  for (int i = 0; i < 8; ++i) {
    int n0 = wave * 128 + i * 16;
    int j = n0 + col;
    kn1v[i] = ws[OFF_KN1 + j];
    km1v[i] = ws[OFF_KM1 + (n0 >> 3) + (col >> 3)];
    #pragma unroll
    for (int v = 0; v < 8; ++v)
      dst[i][v] = ws[OFF_DST + (size_t)(b0 + v + 8 * kh) * 1024 + j];
  }
  __syncthreads();

  const float* Brec = ws + OFF_BP + (size_t)KBI_ * 4096;   // recurrent K-blocks
  const float* bpr  = Brec + (size_t)(wave * 128 + col) * 4 + kh * 2;  // +i*64 +kb*4096
  const float* xproj = ws + OFF_XP;
  int p = 0;

  for (int tl = 0; tl < TC_; ++tl) {
    int t = t0 + tl;

    // ---- accumulators initialized from xproj (d_cur input part + b1) ----
    v8f c[8];
    #pragma unroll
    for (int i = 0; i < 8; ++i) {
      const float* xp = xproj + ((size_t)(tl * B_ + b0 + 8 * kh)) * 1024
                        + wave * 128 + i * 16 + col;
      #pragma unroll
      for (int v = 0; v < 8; ++v) c[i][v] = xp[(size_t)v * 1024];
    }

    // ---- recurrent GEMM: sp1 @ Wrec^T, shared A, double-buffered ----
    const float* ap = &sp1b[p][col * LSP_ + kh * 2];
    v2f a[2], bb[2][8];
    a[0] = *(const v2f*)(ap);
    #pragma unroll
    for (int i = 0; i < 8; ++i) bb[0][i] = *(const v2f*)(bpr + i * 64);
    #pragma unroll
    for (int kb = 0; kb < KBR_; ++kb) {
      int cur = kb & 1, nxt = cur ^ 1;
      if (kb < KBR_ - 1) {
        a[nxt] = *(const v2f*)(ap + (kb + 1) * 4);
        #pragma unroll
        for (int i = 0; i < 8; ++i)
          bb[nxt][i] = *(const v2f*)(bpr + (size_t)(kb + 1) * 4096 + i * 64);
      }
      #pragma unroll
      for (int i = 0; i < 8; ++i) c[i] = wmma_f32(a[cur], bb[cur][i], c[i]);
    }

    // ---- dendritic filter + branch sum + LIF per n-tile ----
    #pragma unroll
    for (int i = 0; i < 8; ++i) {
      int n0 = wave * 128 + i * 16;
      float kn = kn1v[i], km = km1v[i];
      float s[8];
      #pragma unroll
      for (int v = 0; v < 8; ++v) {
        float nd = kn * dst[i][v] + (1.0f - kn) * c[i][v];  // dendritic branch filter
        dst[i][v] = nd;
        nd += __shfl_xor(nd, 1, 32);                        // branch-of-8 sum (wave32)
        nd += __shfl_xor(nd, 2, 32);
        nd += __shfl_xor(nd, 4, 32);
        s[v] = nd;
      }
      if ((col & 7) == 0) {                                 // one writer per 8-lane group
        int nn = (n0 >> 3) + (col >> 3);
        #pragma unroll
        for (int v = 0; v < 8; ++v) {
          int r = v + 8 * kh;
          float m1 = mem1s[r * NH_ + nn];
          float so = sp1b[p][r * LSP_ + nn];
          float mn = m1 * km + (1.0f - km) * s[v] - VTH_ * so;  // LIF + soft reset
          mem1s[r * NH_ + nn] = mn;
          float sn = (mn > VTH_) ? 1.0f : 0.0f;                 // spike_fn(mem1 - VTH)
          sp1b[p ^ 1][r * LSP_ + nn] = sn;
          apres[r * NH_ + nn] += sn;
        }
      }
    }
    __syncthreads();

    // ---- readout: mem2 = k_m2*mem2 + (1-k_m2)*(sp1_new @ W2^T + b2) ----
    for (int idx = tid; idx < 16 * NO_; idx += 256) {
      int r = idx / NO_, o = idx - r * NO_;
      float acc = b2[o];
      const float* sp = &sp1b[p ^ 1][r * LSP_];
      const float* w = &W2s[o * NH_];
      #pragma unroll 8
      for (int n = 0; n < NH_; ++n) acc += sp[n] * w[n];
      float km2 = ws[OFF_KM2 + o];
      float m2 = mem2s[idx] * km2 + (1.0f - km2) * acc;
      mem2s[idx] = m2;
      aposts[idx] += (m2 > 0.0f) ? 1.0f : 0.0f;
    }
    __syncthreads();

    // ---- softmax accumulation for t > 10 (one thread per batch row) ----
    if (tid < 16 && t > 10) {
      float mx = mem2s[tid * NO_];
      for (int o = 1; o < NO_; ++o) mx = fmaxf(mx, mem2s[tid * NO_ + o]);
      float se = 0.0f;
      for (int o = 0; o < NO_; ++o) se += expf(mem2s[tid * NO_ + o] - mx);
      float inv = 1.0f / se;
      for (int o = 0; o < NO_; ++o) outs[tid * NO_ + o] += expf(mem2s[tid * NO_ + o] - mx) * inv;
    }
    __syncthreads();
    p ^= 1;
  }

  // ---- flush persistent state (p == TC_&1 == 1 holds latest sp1) ----
  for (int i = tid; i < 16 * NH_; i += 256) {
    int r = i >> 7, n = i & 127;
    ws[OFF_M1 + (size_t)(b0 + r) * NH_ + n]   = mem1s[i];
    ws[OFF_APRE + (size_t)(b0 + r) * NH_ + n] = apres[i];
    ws[OFF_SP + (size_t)(b0 + r) * NH_ + n]   = sp1b[p][r * LSP_ + n];
  }
  for (int i = tid; i < 16 * NO_; i += 256) {
    int r = i / NO_, o = i - r * NO_;
    ws[OFF_M2 + (size_t)(b0 + r) * NO_ + o]    = mem2s[i];
    ws[OFF_OUT + (size_t)(b0 + r) * NO_ + o]   = outs[i];
    ws[OFF_APOST + (size_t)(b0 + r) * NO_ + o] = aposts[i];
  }
  #pragma unroll
  for (int i = 0; i < 8; ++i) {
    int j = wave * 128 + i * 16 + col;
    #pragma unroll
    for (int v = 0; v < 8; ++v)
      ws[OFF_DST + (size_t)(b0 + v + 8 * kh) * 1024 + j] = dst[i][v];
  }

  // ---- final outputs: [out (256x20) | pre_spike (256x128) | post_spike (256x20)] ----
  if (isLast) {
    for (int i = tid; i < 16 * NO_; i += 256) {
      int r = i / NO_, o = i - r * NO_;
      dout[(size_t)(b0 + r) * NO_ + o] = outs[i];
      dout[(size_t)B_ * NO_ + (size_t)B_ * NH_ + (size_t)(b0 + r) * NO_ + o] =
          (aposts[i] > 0.0f) ? 1.0f : 0.0f;
    }
    for (int i = tid; i < 16 * NH_; i += 256) {
      int r = i >> 7, n = i & 127;
      dout[(size_t)B_ * NO_ + (size_t)(b0 + r) * NH_ + n] = (apres[i] > 0.0f) ? 1.0f : 0.0f;
    }
  }
}

// =====================================================================
extern "C" void kernel_launch(void* const* d_in, const int* in_sizes, int n_in,
                              void* d_out, int out_size, void* d_ws, size_t ws_size,
                              hipStream_t stream) {
  (void)in_sizes; (void)n_in; (void)out_size; (void)ws_size;
  const float* x      = (const float*)d_in[0];
  const float* W1     = (const float*)d_in[1];
  const float* b1     = (const float*)d_in[2];
  const float* tau_m1 = (const float*)d_in[3];
  const float* tau_n1 = (const float*)d_in[4];
  const float* W2     = (const float*)d_in[5];
  const float* b2     = (const float*)d_in[6];
  const float* tau_m2 = (const float*)d_in[7];
  const float* mem0   = (const float*)d_in[8];
  const float* mask   = (const float*)d_in[9];
  float* ws  = (float*)d_ws;
  float* out = (float*)d_out;

  k_init<<<(int)((NBP + 255) / 256), 256, 0, stream>>>(W1, mask, tau_m1, tau_n1, tau_m2, mem0, ws);

  for (int c = 0; c < NCH_; ++c) {
    k_xproj<<<dim3(TC_ * (B_ / 16), 1024 / 256), 256, 0, stream>>>(x, b1, ws, c * TC_);
    k_recur<<<B_ / 16, 256, 0, stream>>>(W2, b2, ws, out, c * TC_, (c == NCH_ - 1) ? 1 : 0);
  }
}